// GCNSampler_43009802502553
// MI455X (gfx1250) — compile-verified
//
#include <hip/hip_runtime.h>

typedef __attribute__((ext_vector_type(2))) float v2f;
typedef __attribute__((ext_vector_type(8))) float v8f;

// ---------------------------------------------------------------------------
// Zero float buffer (float4 granularity; all our sizes are multiples of 4)
// ---------------------------------------------------------------------------
__global__ void zero_kernel(float* __restrict__ p, int n4) {
  int t = blockIdx.x * blockDim.x + threadIdx.x;
  if (t < n4) {
    *(float4*)(p + (size_t)t * 4) = make_float4(0.f, 0.f, 0.f, 0.f);
  }
}

// ---------------------------------------------------------------------------
// Degree accumulation: deg_out[src[e]] += 1, deg_in[dst[e]] += 1
// ---------------------------------------------------------------------------
__global__ void degree_kernel(const int* __restrict__ src,
                              const int* __restrict__ dst,
                              float* __restrict__ deg_out,
                              float* __restrict__ deg_in, int E) {
  int e = blockIdx.x * blockDim.x + threadIdx.x;
  if (e >= E) return;
  atomicAdd(&deg_out[src[e]], 1.0f);
  atomicAdd(&deg_in[dst[e]], 1.0f);
}

// ---------------------------------------------------------------------------
// In-place: deg -> clip(deg,1)^-0.5   (applied to both deg arrays at once)
// ---------------------------------------------------------------------------
__global__ void norm_kernel(float* __restrict__ deg, int n) {
  int t = blockIdx.x * blockDim.x + threadIdx.x;
  if (t >= n) return;
  float d = deg[t];
  d = d < 1.0f ? 1.0f : d;
  deg[t] = 1.0f / sqrtf(d);
}

// ---------------------------------------------------------------------------
// h[node] = x[node] * norm_out[node]   (float4 per thread, 16 threads/node)
// ---------------------------------------------------------------------------
__global__ void scale_kernel(const float* __restrict__ x,
                             const float* __restrict__ norm_out,
                             float* __restrict__ h, int N) {
  int t = blockIdx.x * blockDim.x + threadIdx.x;
  int node = t >> 4;
  if (node >= N) return;
  int g = t & 15;
  float s = norm_out[node];
  float4 v = *(const float4*)(x + (size_t)node * 64 + g * 4);
  v.x *= s; v.y *= s; v.z *= s; v.w *= s;
  *(float4*)(h + (size_t)node * 64 + g * 4) = v;
}

// ---------------------------------------------------------------------------
// Edge scatter: agg[dst[e]] += h[src[e]]  (16 threads/edge, float4 payload,
// fp32 atomics -> L2 atomic units; whole working set is L2-resident)
// ---------------------------------------------------------------------------
__global__ void scatter_kernel(const float* __restrict__ h,
                               const int* __restrict__ src,
                               const int* __restrict__ dst,
                               float* __restrict__ agg, int E) {
  int t = blockIdx.x * blockDim.x + threadIdx.x;
  int e = t >> 4;
  if (e >= E) return;
  int g = t & 15;
  int s = src[e];
  int d = dst[e];
  float4 v = *(const float4*)(h + (size_t)s * 64 + g * 4);
  float* p = agg + (size_t)d * 64 + g * 4;
  atomicAdd(p + 0, v.x);
  atomicAdd(p + 1, v.y);
  atomicAdd(p + 2, v.z);
  atomicAdd(p + 3, v.w);
}

// ---------------------------------------------------------------------------
// Fused GEMM epilogue layer:
//   out = f( (agg * norm_in[row]) @ W + bias )
// MODE==1: f = relu(.) * norm_out[row]  (produces pre-scaled input for the
//          next layer's gather in one pass)
// MODE==0: f = identity (final layer, writes d_out)
//
// One wave32 per 16-row tile. K=64 via 16x V_WMMA_F32_16X16X4_F32 chained
// into a v8f accumulator; 4 N-tiles of 16 columns cover the 64 outputs.
// A-fragment (f32 16x4): lanes 0-15 hold K={4k,4k+1}, lanes 16-31 K={4k+2,4k+3}
// for row M = lane%16 -> one float2 load from the contiguous agg row, scaled
// by norm_in. B-fragment mirrors it (two strided loads of W, L1/L2 resident).
// C/D layout: acc[r] -> row = tile*16 + r + 8*(lane>=16), col = nt*16+lane%16.
// ---------------------------------------------------------------------------
template <int MODE>
__global__ void gcn_gemm_kernel(const float* __restrict__ agg,
                                const float* __restrict__ W,
                                const float* __restrict__ bias,
                                const float* __restrict__ norm_in,
                                const float* __restrict__ norm_out,
                                float* __restrict__ out, int n_tiles) {
  int wave = blockIdx.x * (blockDim.x >> 5) + (threadIdx.x >> 5);
  if (wave >= n_tiles) return;  // uniform per-wave: EXEC stays all-ones
  int lane = threadIdx.x & 31;
  int half = lane >> 4;   // 0: K lo pair, 1: K hi pair
  int l16  = lane & 15;

  int arow = wave * 16 + l16;
  float s_in = norm_in[arow];
  const float* ap = agg + (size_t)arow * 64 + half * 2;

  // Preload all 16 A fragments (32 VGPRs), fold in norm_in scaling.
  v2f afrag[16];
#pragma unroll
  for (int kk = 0; kk < 16; ++kk) {
    float2 t = *(const float2*)(ap + kk * 4);
    afrag[kk].x = t.x * s_in;
    afrag[kk].y = t.y * s_in;
  }

#pragma unroll
  for (int nt = 0; nt < 4; ++nt) {
    int col = nt * 16 + l16;
    v8f acc = {};
#pragma unroll
    for (int kk = 0; kk < 16; ++kk) {
      int krow = kk * 4 + half * 2;
      v2f bfrag;
      bfrag.x = W[(size_t)krow * 64 + col];
      bfrag.y = W[(size_t)(krow + 1) * 64 + col];
      acc = __builtin_amdgcn_wmma_f32_16x16x4_f32(
          /*neg_a=*/false, afrag[kk], /*neg_b=*/false, bfrag,
          /*c_mod=*/(short)0, acc, /*reuse_a=*/false, /*reuse_b=*/false);
    }
    float bv = bias[col];  // col is constant across a lane's 8 accum rows
#pragma unroll
    for (int r = 0; r < 8; ++r) {
      int orow = wave * 16 + half * 8 + r;
      float v = acc[r] + bv;
      if (MODE == 1) {
        v = fmaxf(v, 0.0f);
        v *= norm_out[orow];  // pre-scale for next layer's gather
      }
      out[(size_t)orow * 64 + col] = v;
    }
  }
}

// ---------------------------------------------------------------------------
// Host-side orchestration (graph-capture safe: kernels only, all on stream)
// ---------------------------------------------------------------------------
extern "C" void kernel_launch(void* const* d_in, const int* in_sizes, int n_in,
                              void* d_out, int out_size, void* d_ws,
                              size_t ws_size, hipStream_t stream) {
  const float* x  = (const float*)d_in[0];
  const int*   src = (const int*)d_in[1];
  const int*   dst = (const int*)d_in[2];
  const float* W1 = (const float*)d_in[3];
  const float* b1 = (const float*)d_in[4];
  const float* W2 = (const float*)d_in[5];
  const float* b2 = (const float*)d_in[6];
  float* out = (float*)d_out;

  int N = in_sizes[0] / 64;  // 50000
  int E = in_sizes[1];       // 800000

  // Workspace layout (floats): h[N*64] | agg[N*64] | deg_out[N] | deg_in[N]
  float* ws   = (float*)d_ws;
  float* h    = ws;
  float* agg  = h + (size_t)N * 64;
  float* degO = agg + (size_t)N * 64;  // becomes norm_out in-place
  float* degI = degO + N;              // becomes norm_in in-place

  const int B = 256;
  const int wpb = B / 32;  // waves per block
  int n_tiles = (N + 15) / 16;  // 3125 (N divides exactly)

  // Degrees + norms
  zero_kernel<<<((2 * N / 4) + B - 1) / B, B, 0, stream>>>(degO, 2 * N / 4);
  degree_kernel<<<(E + B - 1) / B, B, 0, stream>>>(src, dst, degO, degI, E);
  norm_kernel<<<(2 * N + B - 1) / B, B, 0, stream>>>(degO, 2 * N);

  // Layer 1: h0 = x * norm_out ; agg = scatter(h0) ; h1s = relu(agg*nin@W1+b1)*norm_out
  scale_kernel<<<(N * 16 + B - 1) / B, B, 0, stream>>>(x, degO, h, N);
  zero_kernel<<<((N * 64 / 4) + B - 1) / B, B, 0, stream>>>(agg, N * 64 / 4);
  scatter_kernel<<<((E * 16) + B - 1) / B, B, 0, stream>>>(h, src, dst, agg, E);
  gcn_gemm_kernel<1><<<(n_tiles + wpb - 1) / wpb, B, 0, stream>>>(
      agg, W1, b1, degI, degO, h, n_tiles);

  // Layer 2: agg = scatter(h1s) ; out = agg*nin @ W2 + b2
  zero_kernel<<<((N * 64 / 4) + B - 1) / B, B, 0, stream>>>(agg, N * 64 / 4);
  scatter_kernel<<<((E * 16) + B - 1) / B, B, 0, stream>>>(h, src, dst, agg, E);
  gcn_gemm_kernel<0><<<(n_tiles + wpb - 1) / wpb, B, 0, stream>>>(
      agg, W2, b2, degI, nullptr, out, n_tiles);
}